// RewardScaling_33543694582021
// MI455X (gfx1250) — compile-verified
//
#include <hip/hip_runtime.h>
#include <cmath>
#include <cstdint>

// Reward scaling: R_t = g*R_{t-1} + r_t ; std_t from running mean/var of R ;
// out_t = r_t / max(std_t, eps).  3-pass decoupled scan:
//   pass1: per-16K-tile aggregates (E, S1, SW, S2) with zero incoming carry
//   pass2: one-block Hillis scan over tile aggregates (double precision)
//   pass3: finalize: per-panel WMMA scans + per-element std + output
// 16-wide scans use V_WMMA_F32_16X16X4_F32: D = L(16x16)*X(16x16), L lower-
// triangular (gamma powers or ones), K-accumulated 4x (K=4).  Cross-chunk
// carries use DPP16 row_shr log-scans (pure VALU); panels are double-buffered
// through LDS with global_load_async_to_lds_b128 / s_wait_asynccnt.

#define GAMMA  0.99f
#define EPSV   1e-8f
#define PANEL  256        // elements per WMMA panel (one wave)
#define TILE   16384      // elements per block
#define WAVES  8
#define PPW    8          // panels per wave
#define NPAN   64         // panels per tile
#define BLOCK  256

typedef float v2f __attribute__((ext_vector_type(2)));
typedef float v4f __attribute__((ext_vector_type(4)));
typedef float v8f __attribute__((ext_vector_type(8)));

// Low 32 bits of a generic LDS address are the LDS byte offset (ISA 10.2).
static __device__ __forceinline__ unsigned lds_off(const void* p) {
  return (unsigned)(uintptr_t)p;
}

// Issue (no wait): 256 contiguous floats (1 KB) global -> this wave's LDS
// region via the CDNA5 async-to-LDS path (ASYNCcnt).
static __device__ __forceinline__ void stage_issue(const float* gbase,
                                                   unsigned gByte,
                                                   unsigned ldsByte) {
  asm volatile(
      "global_load_async_to_lds_b128 %0, %1, %4\n\t"
      "global_load_async_to_lds_b128 %2, %3, %4"
      :
      : "v"(ldsByte), "v"(gByte), "v"(ldsByte + 512u), "v"(gByte + 512u),
        "s"(gbase)
      : "memory");
}
// Wait until only the most recent panel's 2 async loads may remain in flight.
static __device__ __forceinline__ void stage_wait_prev() {
  asm volatile("s_wait_asynccnt 2" ::: "memory");
}
static __device__ __forceinline__ void stage_wait_all() {
  asm volatile("s_wait_asynccnt 0" ::: "memory");
}

// v_mov_b32_dpp row_shr:S with zero fill at the row (16-lane) boundary.
template <int CTRL>
static __device__ __forceinline__ float dpp_shr0(float x) {
  return __int_as_float(__builtin_amdgcn_update_dpp(
      0, __float_as_int(x), CTRL, 0xf, 0xf, false));
}

// Inclusive weighted row scan: y_n = sum_{m<=n} g^{n-m} y_m within each
// 16-lane row, log-depth, pure VALU (DPP16 row_shr, zeros shifted in).
static __device__ __forceinline__ float row_scan_weighted(float y, float g) {
  const float g2 = g * g, g4 = g2 * g2, g8 = g4 * g4;
  y = __fmaf_rn(dpp_shr0<0x111>(y), g, y);   // row_shr:1
  y = __fmaf_rn(dpp_shr0<0x112>(y), g2, y);  // row_shr:2
  y = __fmaf_rn(dpp_shr0<0x114>(y), g4, y);  // row_shr:4
  y = __fmaf_rn(dpp_shr0<0x118>(y), g8, y);  // row_shr:8
  return y;
}

// Chunk m's end value lives in lane 16+m (M=15 row of the WMMA D matrix).
// Broadcast to both halves, then carry into chunk n = exclusive weighted scan.
static __device__ __forceinline__ float chunk_carry(float endv, int h,
                                                    float g16) {
  float other = __shfl_xor(endv, 16, 32);
  float Em = h ? endv : other;                   // E_{lane&15} on all lanes
  float incl = row_scan_weighted(Em, g16);       // incl_n
  return dpp_shr0<0x111>(incl);                  // C_n = incl_{n-1}, C_0 = 0
}
// Exclusive prefix of chunk totals (ones-scan cross-chunk offset).
static __device__ __forceinline__ float chunk_prefix(float tot, int h) {
  float other = __shfl_xor(tot, 16, 32);
  float Tm = h ? tot : other;
  float incl = row_scan_weighted(Tm, 1.0f);
  return dpp_shr0<0x111>(incl);
}

// Panel-local (zero-carry) scan R_t = sum_{j<=t} g^{t-j} r_j over 256 elems
// already resident in LDS (stg).  X is column-major 16x16 (column n = elems
// 16n..16n+15).  Operand layouts (ISA 7.12.2): A 16x4: vgpr v, lanes<16 ->
// K=v, lanes>=16 -> K=v+2.  B 4x16 mirrors it: vgpr v, lanes<16 -> row v
// (N=lane), lanes>=16 -> row v+2.  C/D: vgpr p, lanes<16 -> M=p, lanes>=16 ->
// M=p+8  => lane holds D elements 16*(lane&15) + 8*(lane>>4) + p (contiguous).
static __device__ __forceinline__ void gamma_scan_resident(
    const float* stg, const v2f aG[4], int n_, int h, float g16, float wbase,
    float Rloc[8]) {
  v8f acc = {};
#pragma unroll
  for (int k = 0; k < 4; ++k) {
    v2f b = *(const v2f*)&stg[16 * n_ + 4 * k + 2 * h];
    acc = __builtin_amdgcn_wmma_f32_16x16x4_f32(false, aG[k], false, b,
                                                (short)0, acc, false, false);
  }
  const float C = chunk_carry(acc[7], h, g16);
  float w = wbase;  // gamma^{k+1}, k = 8h..8h+7
#pragma unroll
  for (int p = 0; p < 8; ++p) {
    Rloc[p] = acc[p] + C * w;
    w *= GAMMA;
  }
}

// ---------------------------------------------------------------- pass 1 ----
__global__ void __launch_bounds__(BLOCK) k_pass1(
    const float* __restrict__ rewards, float4* __restrict__ tileAgg) {
  __shared__ __align__(16) float stage[WAVES * 2 * PANEL];  // double buffer
  __shared__ float4 ldsAgg[NPAN];

  const int tid = threadIdx.x, wave = tid >> 5, lane = tid & 31;
  const int n_ = lane & 15, h = lane >> 4;

  v2f aG[4];  // A-operand slices of L_gamma
#pragma unroll
  for (int k = 0; k < 4; ++k) {
    int j0 = 4 * k + 2 * h;
    aG[k].x = (j0 <= n_) ? __powf(GAMMA, (float)(n_ - j0)) : 0.f;
    aG[k].y = (j0 + 1 <= n_) ? __powf(GAMMA, (float)(n_ - j0 - 1)) : 0.f;
  }
  const float g16 = __powf(GAMMA, 16.f);
  const float wbase = __powf(GAMMA, (float)(8 * h + 1));
  const float gw = __powf(GAMMA, (float)(16 * n_));

  float* stg0 = &stage[wave * 2 * PANEL];
  const unsigned tileBase = (unsigned)blockIdx.x * TILE;
  const unsigned waveBase = tileBase + (unsigned)(wave * PPW) * PANEL;

  stage_issue(rewards, waveBase * 4u + 16u * lane, lds_off(stg0) + 16u * lane);
  for (int it = 0; it < PPW; ++it) {
    const int pi = wave * PPW + it;
    const unsigned base = tileBase + (unsigned)pi * PANEL;
    if (it + 2 < PPW)
      __builtin_prefetch(rewards + base + 2 * PANEL + lane * 8, 0, 1);
    if (it + 1 < PPW) {
      float* nxt = stg0 + ((it + 1) & 1) * PANEL;
      stage_issue(rewards, (base + PANEL) * 4u + 16u * lane,
                  lds_off(nxt) + 16u * lane);
      stage_wait_prev();
    } else {
      stage_wait_all();
    }
    const float* stg = stg0 + (it & 1) * PANEL;
    float Rloc[8];
    gamma_scan_resident(stg, aG, n_, h, g16, wbase, Rloc);
    // panel aggregates: e, s1 = sum R, sw = sum g^{tau+1} R, s2 = sum R^2
    float s1 = 0.f, s2 = 0.f, sw = 0.f, w = wbase;
#pragma unroll
    for (int p = 0; p < 8; ++p) {
      float r = Rloc[p];
      s1 += r; s2 += r * r; sw += gw * w * r; w *= GAMMA;
    }
#pragma unroll
    for (int off = 16; off; off >>= 1) {
      s1 += __shfl_xor(s1, off, 32);
      s2 += __shfl_xor(s2, off, 32);
      sw += __shfl_xor(sw, off, 32);
    }
    float e = __shfl(Rloc[7], 31, 32);  // element 255 of panel
    if (lane == 0) ldsAgg[pi] = make_float4(e, s1, sw, s2);
  }
  __syncthreads();
  if (tid == 0) {  // combine 64 panel aggregates -> tile aggregate
    const float Wp = __powf(GAMMA, (float)PANEL);
    const float G1p = GAMMA * (1.f - Wp) / (1.f - GAMMA);
    const float G2p = GAMMA * GAMMA * (1.f - Wp * Wp) / (1.f - GAMMA * GAMMA);
    float E = 0.f, S1 = 0.f, SW = 0.f, S2 = 0.f, gl = 1.f;
    for (int p = 0; p < NPAN; ++p) {
      float4 a = ldsAgg[p];
      float c = E;
      S1 += a.y + c * G1p;
      S2 += a.w + 2.f * c * a.z + c * c * G2p;
      SW += gl * (a.z + c * G2p);
      E = a.x + c * Wp;
      gl *= Wp;
    }
    tileAgg[blockIdx.x] = make_float4(E, S1, SW, S2);
  }
}

// ---------------------------------------------------------------- pass 2 ----
struct Seg { double E, S1, SW, S2, W, G1, G2; };

static __device__ __forceinline__ Seg seg_combine(const Seg& L, const Seg& R) {
  Seg o;
  double c = L.E;
  o.E = R.E + c * R.W;
  o.S1 = L.S1 + R.S1 + c * R.G1;
  o.SW = L.SW + L.W * (R.SW + c * R.G2);
  o.S2 = L.S2 + R.S2 + 2.0 * c * R.SW + c * c * R.G2;
  o.W = L.W * R.W;
  o.G1 = L.G1 + L.W * R.G1;
  o.G2 = L.G2 + L.W * L.W * R.G2;
  return o;
}

__global__ void __launch_bounds__(1024) k_pass2(
    const float4* __restrict__ tileAgg, double* __restrict__ tilePre,
    int nTiles) {
  __shared__ Seg s[1024];
  const int t = threadIdx.x;
  Seg v;
  if (t < nTiles) {
    float4 a = tileAgg[t];
    const double G = 0.99;
    const double W = pow(G, (double)TILE);
    v.E = a.x; v.S1 = a.y; v.SW = a.z; v.S2 = a.w;
    v.W = W;
    v.G1 = G * (1.0 - W) / (1.0 - G);
    v.G2 = G * G * (1.0 - W * W) / (1.0 - G * G);
  } else {
    v.E = 0; v.S1 = 0; v.SW = 0; v.S2 = 0; v.W = 1.0; v.G1 = 0; v.G2 = 0;
  }
  s[t] = v;
  __syncthreads();
  for (int off = 1; off < 1024; off <<= 1) {  // Hillis inclusive scan
    Seg L;
    const bool has = (t >= off);
    if (has) L = s[t - off];
    __syncthreads();
    if (has) s[t] = seg_combine(L, s[t]);
    __syncthreads();
  }
  if (t < nTiles) {  // exclusive prefix per tile: carry-in, S1/S2 offsets
    double E = 0.0, S1 = 0.0, S2 = 0.0;
    if (t > 0) { E = s[t - 1].E; S1 = s[t - 1].S1; S2 = s[t - 1].S2; }
    tilePre[3 * t + 0] = E;
    tilePre[3 * t + 1] = S1;
    tilePre[3 * t + 2] = S2;
  }
}

// ---------------------------------------------------------------- pass 3 ----
__global__ void __launch_bounds__(BLOCK) k_pass3(
    const float* __restrict__ rewards, float* __restrict__ out,
    const double* __restrict__ tilePre) {
  __shared__ __align__(16) float stage[WAVES * 2 * PANEL];
  __shared__ float4 ldsAgg[NPAN];
  __shared__ double ldsCar[NPAN], ldsS1[NPAN], ldsS2[NPAN];

  const int tid = threadIdx.x, wave = tid >> 5, lane = tid & 31;
  const int n_ = lane & 15, h = lane >> 4;

  v2f aG[4], aO[4];  // gamma-triangular and ones-triangular A operands
#pragma unroll
  for (int k = 0; k < 4; ++k) {
    int j0 = 4 * k + 2 * h;
    aG[k].x = (j0 <= n_) ? __powf(GAMMA, (float)(n_ - j0)) : 0.f;
    aG[k].y = (j0 + 1 <= n_) ? __powf(GAMMA, (float)(n_ - j0 - 1)) : 0.f;
    aO[k].x = (j0 <= n_) ? 1.f : 0.f;
    aO[k].y = (j0 + 1 <= n_) ? 1.f : 0.f;
  }
  const float g16 = __powf(GAMMA, 16.f);
  const float wbase = __powf(GAMMA, (float)(8 * h + 1));
  const float gw = __powf(GAMMA, (float)(16 * n_));

  float* stg0 = &stage[wave * 2 * PANEL];
  const unsigned tileBase = (unsigned)blockIdx.x * TILE;
  const unsigned waveBase = tileBase + (unsigned)(wave * PPW) * PANEL;

  // phase 1: per-panel aggregates (same math as pass 1)
  stage_issue(rewards, waveBase * 4u + 16u * lane, lds_off(stg0) + 16u * lane);
  for (int it = 0; it < PPW; ++it) {
    const int pi = wave * PPW + it;
    const unsigned base = tileBase + (unsigned)pi * PANEL;
    if (it + 1 < PPW) {
      float* nxt = stg0 + ((it + 1) & 1) * PANEL;
      stage_issue(rewards, (base + PANEL) * 4u + 16u * lane,
                  lds_off(nxt) + 16u * lane);
      stage_wait_prev();
    } else {
      stage_wait_all();
    }
    const float* stg = stg0 + (it & 1) * PANEL;
    float Rloc[8];
    gamma_scan_resident(stg, aG, n_, h, g16, wbase, Rloc);
    float s1 = 0.f, s2 = 0.f, sw = 0.f, w = wbase;
#pragma unroll
    for (int p = 0; p < 8; ++p) {
      float r = Rloc[p];
      s1 += r; s2 += r * r; sw += gw * w * r; w *= GAMMA;
    }
#pragma unroll
    for (int off = 16; off; off >>= 1) {
      s1 += __shfl_xor(s1, off, 32);
      s2 += __shfl_xor(s2, off, 32);
      sw += __shfl_xor(sw, off, 32);
    }
    float e = __shfl(Rloc[7], 31, 32);
    if (lane == 0) ldsAgg[pi] = make_float4(e, s1, sw, s2);
  }
  __syncthreads();

  // phase 2: sequential 64-panel scan seeded by the tile's global prefix
  if (tid == 0) {
    const double Wp = pow(0.99, (double)PANEL);
    const double G1p = 0.99 * (1.0 - Wp) / (1.0 - 0.99);
    const double G2p = 0.99 * 0.99 * (1.0 - Wp * Wp) / (1.0 - 0.99 * 0.99);
    double E = tilePre[3 * blockIdx.x + 0];
    double S1 = tilePre[3 * blockIdx.x + 1];
    double S2 = tilePre[3 * blockIdx.x + 2];
    for (int p = 0; p < NPAN; ++p) {
      float4 a = ldsAgg[p];
      ldsCar[p] = E; ldsS1[p] = S1; ldsS2[p] = S2;
      double c = E;
      S1 += (double)a.y + c * G1p;
      S2 += (double)a.w + 2.0 * c * (double)a.z + c * c * G2p;
      E = (double)a.x + c * Wp;
    }
  }
  __syncthreads();

  // phase 3: recompute panel R, add carry, ones-WMMA prefix sums, emit output
  stage_issue(rewards, waveBase * 4u + 16u * lane, lds_off(stg0) + 16u * lane);
  for (int it = 0; it < PPW; ++it) {
    const int pi = wave * PPW + it;
    const unsigned base = tileBase + (unsigned)pi * PANEL;
    if (it + 1 < PPW) {
      float* nxt = stg0 + ((it + 1) & 1) * PANEL;
      stage_issue(rewards, (base + PANEL) * 4u + 16u * lane,
                  lds_off(nxt) + 16u * lane);
      stage_wait_prev();
    } else {
      stage_wait_all();
    }
    float* stg = stg0 + (it & 1) * PANEL;
    float Rloc[8];
    gamma_scan_resident(stg, aG, n_, h, g16, wbase, Rloc);
    const float cp = (float)ldsCar[pi];
    const double S1o = ldsS1[pi], S2o = ldsS2[pi];
    // publish Rloc back into this panel's staging buffer in element order
    v4f lo, hi;
    lo.x = Rloc[0]; lo.y = Rloc[1]; lo.z = Rloc[2]; lo.w = Rloc[3];
    hi.x = Rloc[4]; hi.y = Rloc[5]; hi.z = Rloc[6]; hi.w = Rloc[7];
    *(v4f*)&stg[16 * n_ + 8 * h] = lo;
    *(v4f*)&stg[16 * n_ + 8 * h + 4] = hi;
    __builtin_amdgcn_wave_barrier();
    // B-layout reload, add panel carry, square; two ones-triangular scans
    v8f p1 = {}, p2 = {};
#pragma unroll
    for (int k = 0; k < 4; ++k) {
      v2f v = *(const v2f*)&stg[16 * n_ + 4 * k + 2 * h];
      float w0 = __powf(GAMMA, (float)(16 * n_ + 4 * k + 2 * h + 1));
      v2f r1, r2;
      r1.x = v.x + cp * w0;
      r1.y = v.y + cp * w0 * GAMMA;
      r2.x = r1.x * r1.x;
      r2.y = r1.y * r1.y;
      p1 = __builtin_amdgcn_wmma_f32_16x16x4_f32(false, aO[k], false, r1,
                                                 (short)0, p1, false, false);
      p2 = __builtin_amdgcn_wmma_f32_16x16x4_f32(false, aO[k], false, r2,
                                                 (short)0, p2, false, false);
    }
    const float o1 = chunk_prefix(p1[7], h);
    const float o2 = chunk_prefix(p2[7], h);
    // reload r (L2-resident) and compute outputs
    const float* rp = rewards + base + 16 * n_ + 8 * h;
    v4f ra = *(const v4f*)rp;
    v4f rb = *(const v4f*)(rp + 4);
    float rv[8] = {ra.x, ra.y, ra.z, ra.w, rb.x, rb.y, rb.z, rb.w};
    float res[8];
    const long long gb = (long long)base + 16 * n_ + 8 * h;
#pragma unroll
    for (int p = 0; p < 8; ++p) {
      const long long g = gb + p;
      const double cnt = (double)(g + 1);
      const double S1v = S1o + (double)(o1 + p1[p]);
      const double S2v = S2o + (double)(o2 + p2[p]);
      double var = S2v - S1v * S1v / cnt;
      float sd;
      if (g == 0) sd = 1.f;
      else { if (var < 0.0) var = 0.0; sd = (float)sqrt(var / cnt); }
      sd = fmaxf(sd, EPSV);
      res[p] = rv[p] / sd;
    }
    v4f olo, ohi;
    olo.x = res[0]; olo.y = res[1]; olo.z = res[2]; olo.w = res[3];
    ohi.x = res[4]; ohi.y = res[5]; ohi.z = res[6]; ohi.w = res[7];
    float* op = out + base + 16 * n_ + 8 * h;
    *(v4f*)op = olo;
    *(v4f*)(op + 4) = ohi;
    __builtin_amdgcn_wave_barrier();
  }
}

// ------------------------------------------------------------- launcher ----
extern "C" void kernel_launch(void* const* d_in, const int* in_sizes, int n_in,
                              void* d_out, int out_size, void* d_ws,
                              size_t ws_size, hipStream_t stream) {
  (void)n_in; (void)out_size; (void)ws_size;
  const float* rewards = (const float*)d_in[0];
  float* out = (float*)d_out;
  const int N = in_sizes[0];             // expected 16777216 (multiple of TILE)
  const int nTiles = (N + TILE - 1) / TILE;  // expected 1024 (<= 1024)

  char* ws = (char*)d_ws;
  float4* tileAgg = (float4*)ws;                 // nTiles * 16 B
  double* tilePre = (double*)(ws + 32768);       // nTiles * 24 B

  k_pass1<<<nTiles, BLOCK, 0, stream>>>(rewards, tileAgg);
  k_pass2<<<1, 1024, 0, stream>>>(tileAgg, tilePre, nTiles);
  k_pass3<<<nTiles, BLOCK, 0, stream>>>(rewards, out, tilePre);
}